// NutritionLoss_26465588478297
// MI455X (gfx1250) — compile-verified
//
#include <hip/hip_runtime.h>

typedef __attribute__((ext_vector_type(2))) float v2f;
typedef __attribute__((ext_vector_type(8))) float v8f;

#define NUM_FOODS 223
#define NCOL 5
#define ENTRIES_PER_BATCH 420   // 7 days * 5 meals * 12 foods
#define BATCH 512

// Kernel 1: one workgroup per batch element.
// Gathers data[id,:] from an LDS-staged table, accumulates amount-weighted
// 5-vectors for gold and pred, reduces across the block, emits
// s_b = sum_c |pred_c - gold_c| / 700  into d_ws[b].
__global__ __launch_bounds__(128) void nutrition_batch_kernel(
    const float* __restrict__ y_pred,   // [512,420,2] (id, amount)
    const float* __restrict__ y,        // [512,420,2]
    const float* __restrict__ data,     // [223,5]
    float* __restrict__ ws)             // [512] per-batch L1 partials
{
    __shared__ float sdata[NUM_FOODS * NCOL];   // 4460 B food table
    __shared__ float wred[4][2][NCOL];          // per-wave partials

    const int tid = threadIdx.x;
    const int b   = blockIdx.x;

    // Stage the nutrition table into LDS (shared by all gathers in this WGP).
    for (int i = tid; i < NUM_FOODS * NCOL; i += 128) sdata[i] = data[i];
    __syncthreads();

    float gg[NCOL] = {0.f, 0.f, 0.f, 0.f, 0.f};  // gold accumulators
    float gp[NCOL] = {0.f, 0.f, 0.f, 0.f, 0.f};  // pred accumulators

    const float2* ty = (const float2*)y      + (size_t)b * ENTRIES_PER_BATCH;
    const float2* tp = (const float2*)y_pred + (size_t)b * ENTRIES_PER_BATCH;

    for (int i = tid; i < ENTRIES_PER_BATCH; i += 128) {
        // gold: ids are exact integer floats 0..222
        float2 t = ty[i];
        int idt = (int)rintf(t.x);
        const float* rowt = &sdata[idt * NCOL];
        #pragma unroll
        for (int c = 0; c < NCOL; ++c) gg[c] = fmaf(t.y, rowt[c], gg[c]);

        // pred: round half-to-even (matches jnp.round); _bound maps id>222 -> ~0 -> row 0
        float2 p = tp[i];
        int idp = (int)rintf(p.x);
        if ((unsigned)idp > 222u) idp = 0;
        const float* rowp = &sdata[idp * NCOL];
        #pragma unroll
        for (int c = 0; c < NCOL; ++c) gp[c] = fmaf(p.y, rowp[c], gp[c]);
    }

    // wave32 butterfly reduction of the 10 accumulators
    #pragma unroll
    for (int c = 0; c < NCOL; ++c) {
        #pragma unroll
        for (int m = 16; m > 0; m >>= 1) {
            gg[c] += __shfl_xor(gg[c], m, 32);
            gp[c] += __shfl_xor(gp[c], m, 32);
        }
    }

    const int wave = tid >> 5;
    const int lane = tid & 31;
    if (lane == 0) {
        #pragma unroll
        for (int c = 0; c < NCOL; ++c) {
            wred[wave][0][c] = gg[c];
            wred[wave][1][c] = gp[c];
        }
    }
    __syncthreads();

    if (tid == 0) {
        float s = 0.f;
        #pragma unroll
        for (int c = 0; c < NCOL; ++c) {
            float G = wred[0][0][c] + wred[1][0][c] + wred[2][0][c] + wred[3][0][c];
            float P = wred[0][1][c] + wred[1][1][c] + wred[2][1][c] + wred[3][1][c];
            s += fabsf(P - G);
        }
        // fold in the /100 (amounts) and /7 (days) scaling once per batch
        ws[b] = s * (1.0f / 700.0f);
    }
}

// Kernel 2: reduce the 512 per-batch partials with V_WMMA_F32_16X16X4_F32.
// B = all-ones, so D[m,n] = rowsum(A[m,:]) for every column n; accumulating
// 8 tiles of 16x4 f32 through C reduces all 512 values in one wave.
// Lane p holds sum(rows 0..7) in its 8 D VGPRs for p<16 and sum(rows 8..15)
// for p>=16, so partial + shfl_xor(partial,16) is the grand total in every lane.
__global__ __launch_bounds__(32) void reduce512_wmma_kernel(
    const float* __restrict__ ws, float* __restrict__ out)
{
    const int lane = threadIdx.x;   // exactly one wave32; EXEC all-ones at WMMA

    v2f bones;
    bones[0] = 1.0f;
    bones[1] = 1.0f;

    v8f c = {};
    #pragma unroll
    for (int i = 0; i < 8; ++i) {
        v2f a;
        a[0] = ws[i * 64 + lane];
        a[1] = ws[i * 64 + 32 + lane];
        // 8 args: (neg_a, A, neg_b, B, c_mod, C, reuse_a, reuse_b)
        c = __builtin_amdgcn_wmma_f32_16x16x4_f32(
            false, a, false, bones, (short)0, c, false, false);
    }

    float partial = c[0] + c[1] + c[2] + c[3] + c[4] + c[5] + c[6] + c[7];
    float total = partial + __shfl_xor(partial, 16, 32);

    if (lane == 0) out[0] = total * (5.0f / 512.0f);   // NUTRITION_PENALTY * mean over batch
}

extern "C" void kernel_launch(void* const* d_in, const int* in_sizes, int n_in,
                              void* d_out, int out_size, void* d_ws, size_t ws_size,
                              hipStream_t stream) {
    const float* y_pred = (const float*)d_in[0];  // [512,7,5,12,2]
    const float* y      = (const float*)d_in[1];  // [512,7,5,12,2]
    const float* data   = (const float*)d_in[2];  // [223,5]
    float* ws  = (float*)d_ws;                    // 512 floats of scratch
    float* out = (float*)d_out;                   // scalar loss

    nutrition_batch_kernel<<<BATCH, 128, 0, stream>>>(y_pred, y, data, ws);
    reduce512_wmma_kernel<<<1, 32, 0, stream>>>(ws, out);
}